// HF_19670950215891
// MI455X (gfx1250) — compile-verified
//
#include <hip/hip_runtime.h>

// Householder rank-1 apply:  out[b,:] = s[b,:] - (2*<v,s>/<v,v>) * v[b,:]
// B = 512 rows, K = 512 cols, fp32 in / fp32 out.
//
// Memory-bound: 3 MB total traffic -> ~130 ns at 23.3 TB/s. No WMMA use is
// justified (rank-1 update; a WMMA formulation wastes 15/16 of each tile or
// re-materializes H at 3000x the FLOPs). CDNA5 path used instead: async
// global->LDS copies (ASYNCcnt) + wave32 shuffle reductions.

constexpr int KDIM = 512;
constexpr int TPB  = 128;   // 4 waves per block; each thread owns 4 floats (16B)

__global__ __launch_bounds__(TPB) void hh_rank1_kernel(
    const float* __restrict__ v,
    const float* __restrict__ s,
    float* __restrict__ out)
{
    __shared__ float smV[KDIM];
    __shared__ float smS[KDIM];
    __shared__ float redVS[TPB / 32];
    __shared__ float redVV[TPB / 32];

    const int row  = blockIdx.x;
    const int t    = threadIdx.x;
    const int lane = t & 31;
    const int wave = t >> 5;

    const float* vrow = v + (size_t)row * KDIM;   // uniform -> SGPR pair
    const float* srow = s + (size_t)row * KDIM;

    // Per-lane byte offset into the row, and per-lane LDS destination byte
    // address (addrspace(3) ptr -> 32-bit LDS offset).
    const unsigned byteOff = (unsigned)t * 16u;
    const unsigned ldsV = (unsigned)(size_t)(&smV[t * 4]);
    const unsigned ldsS = (unsigned)(size_t)(&smS[t * 4]);

    // gfx1250 async copy: copies 16B per lane from global base saddr plus offset vaddr to the LDS address in vdst.
    asm volatile("global_load_async_to_lds_b128 %0, %1, %2"
                 :: "v"(ldsV), "v"(byteOff), "s"(vrow) : "memory");
    asm volatile("global_load_async_to_lds_b128 %0, %1, %2"
                 :: "v"(ldsS), "v"(byteOff), "s"(srow) : "memory");
    // Per-wave counter; each thread only reads back its own lane's bytes,
    // so no cross-wave barrier is needed before the LDS reads.
    asm volatile("s_wait_asynccnt 0x0" ::: "memory");

    const float4 v4 = *reinterpret_cast<const float4*>(&smV[t * 4]);
    const float4 s4 = *reinterpret_cast<const float4*>(&smS[t * 4]);

    float pvs = v4.x * s4.x + v4.y * s4.y + v4.z * s4.z + v4.w * s4.w;
    float pvv = v4.x * v4.x + v4.y * v4.y + v4.z * v4.z + v4.w * v4.w;

    // wave32 butterfly reduction
    #pragma unroll
    for (int m = 16; m >= 1; m >>= 1) {
        pvs += __shfl_xor(pvs, m, 32);
        pvv += __shfl_xor(pvv, m, 32);
    }
    if (lane == 0) { redVS[wave] = pvs; redVV[wave] = pvv; }
    __syncthreads();

    const float dvs = redVS[0] + redVS[1] + redVS[2] + redVS[3];
    const float dvv = redVV[0] + redVV[1] + redVV[2] + redVV[3];
    const float scale = 2.0f * dvs / dvv;

    float4 o4;
    o4.x = s4.x - scale * v4.x;
    o4.y = s4.y - scale * v4.y;
    o4.z = s4.z - scale * v4.z;
    o4.w = s4.w - scale * v4.w;
    *reinterpret_cast<float4*>(out + (size_t)row * KDIM + t * 4) = o4;
}

extern "C" void kernel_launch(void* const* d_in, const int* in_sizes, int n_in,
                              void* d_out, int out_size, void* d_ws, size_t ws_size,
                              hipStream_t stream) {
    // inputs: d_in[0] = i (unused scalar), d_in[1] = v [B,K], d_in[2] = s [B,K]
    const float* v = (const float*)d_in[1];
    const float* s = (const float*)d_in[2];
    float* out = (float*)d_out;

    const int B = in_sizes[1] / KDIM;   // 512
    hh_rank1_kernel<<<dim3(B), dim3(TPB), 0, stream>>>(v, s, out);
}